// ImplicitCell_26027501813784
// MI455X (gfx1250) — compile-verified
//
#include <hip/hip_runtime.h>
#include <math.h>

#define ALPHA_C   0.5f
#define MAX_IT    100
#define NV        64
#define STR       68            // padded LDS row stride (floats): bank-conflict-free, 16B-aligned rows
#define NEWTON_IT 5             // err0 <= ~0.2 -> err^(2^4) ~ 1e-14 already below fp32; 5 = margin

#if defined(__HIP_DEVICE_COMPILE__) && __has_builtin(__builtin_amdgcn_global_load_async_to_lds_b128) && __has_builtin(__builtin_amdgcn_s_wait_asynccnt)
#define USE_ASYNC_LDS 1
#else
#define USE_ASYNC_LDS 0
#endif

typedef __attribute__((ext_vector_type(2))) float v2f;
typedef __attribute__((ext_vector_type(8))) float v8f;

#if USE_ASYNC_LDS
// builtin wants 16B int-vector pointers (per hipcc diagnostic): AS1 source, AS3 dest
typedef int v4i_vs __attribute__((vector_size(16)));
typedef __attribute__((address_space(1))) v4i_vs* g128_t;
typedef __attribute__((address_space(3))) v4i_vs* l128_t;
#endif

// V_WMMA_F32_16X16X4_F32: A=16x4 f32 (2 VGPR), B=4x16 f32 (2 VGPR), C/D=16x16 f32 (8 VGPR)
__device__ __forceinline__ v8f wmma_f32_16x16x4(v2f a, v2f b, v8f c) {
  return __builtin_amdgcn_wmma_f32_16x16x4_f32(false, a, false, b, (short)0, c, false, false);
}

// A fragment (16x4): lanes 0-15 -> M=lane, K=0/1 in v0/v1 ; lanes 16-31 -> K=2/3
__device__ __forceinline__ v2f lds_afrag(const float* W, int mrow, int k4, int lane) {
  const int r = mrow + (lane & 15);
  const int c = k4 + ((lane >> 4) << 1);
  v2f a;
  a.x = W[r * STR + c];
  a.y = W[r * STR + c + 1];
  return a;
}

// B fragment (4x16): v0 = rows K=0 (lanes 0-15) / K=2 (lanes 16-31); v1 = rows K=1 / K=3
__device__ __forceinline__ v2f lds_bfrag(const float* W, int k4, int ncol, int lane) {
  const int n  = ncol + (lane & 15);
  const int kr = k4 + ((lane >> 4) << 1);
  v2f b;
  b.x = W[kr * STR + n];
  b.y = W[(kr + 1) * STR + n];
  return b;
}

// C/D tile (16x16): VGPR v, lanes 0-15 -> (M=v, N=lane); lanes 16-31 -> (M=v+8, N=lane-16)
__device__ __forceinline__ void lds_store_ctile(float* W, int mrow, int ncol, int lane, v8f c) {
  const int n  = ncol + (lane & 15);
  const int r0 = mrow + ((lane >> 4) << 3);
#pragma unroll
  for (int v = 0; v < 8; ++v) W[(r0 + v) * STR + n] = c[v];
}

__global__ __launch_bounds__(32)
void implicit_cell_pr_kernel(const float* __restrict__ D11,
                             const float* __restrict__ bias,
                             const float* __restrict__ x_init,
                             float* __restrict__ out) {
  __shared__ float Wm[NV * STR];   // M = 1.5 I - 0.5 D11[b]
  __shared__ float Wx[NV * STR];   // Newton iterate -> M^-1
  __shared__ float Wt[NV * STR];   // staging for D11, then T = M * X
  __shared__ float ab[NV], xv[NV], uh[NV], vv[NV], yn[NV];

  const int b    = blockIdx.x;
  const int lane = threadIdx.x;
  const float* Db = D11 + (size_t)b * NV * NV;

#if USE_ASYNC_LDS
  // ---- async copy raw D11[b] tile into LDS staging (ASYNCcnt path, no VGPR round-trip) ----
  for (int rr = 0; rr < NV; rr += 2) {
    const int r = rr + (lane >> 4);          // lanes 0-15 row rr, lanes 16-31 row rr+1
    const int c = (lane & 15) << 2;          // 16B per lane, coalesced 512B per instruction
    __builtin_amdgcn_global_load_async_to_lds_b128(
        (g128_t)(Db + r * NV + c),
        (l128_t)(&Wt[r * STR + c]),
        0, 0);
  }
  __builtin_amdgcn_s_wait_asynccnt(0);
  // form M = 1.5 I - 0.5 D11 and X0 = (2/3) I from the staged tile
  for (int rr = 0; rr < NV; rr += 2) {
    const int r = rr + (lane >> 4);
    const int c = (lane & 15) << 2;
    const float4 d = *(const float4*)(&Wt[r * STR + c]);
    const float dv[4] = {d.x, d.y, d.z, d.w};
#pragma unroll
    for (int j = 0; j < 4; ++j) {
      Wm[r * STR + c + j] = ((r == c + j) ? 1.5f : 0.0f) - 0.5f * dv[j];
      Wx[r * STR + c + j] = (r == c + j) ? (2.0f / 3.0f) : 0.0f;
    }
  }
#else
  // ---- fallback: direct coalesced load ----
  for (int rr = 0; rr < NV; rr += 2) {
    const int r = rr + (lane >> 4);
    const int c = (lane & 15) << 2;
    const float4 d = *(const float4*)(Db + r * NV + c);
    const float dv[4] = {d.x, d.y, d.z, d.w};
#pragma unroll
    for (int j = 0; j < 4; ++j) {
      Wm[r * STR + c + j] = ((r == c + j) ? 1.5f : 0.0f) - 0.5f * dv[j];
      Wx[r * STR + c + j] = (r == c + j) ? (2.0f / 3.0f) : 0.0f;
    }
  }
#endif
  ab[lane]      = ALPHA_C * bias[(size_t)b * NV + lane];
  ab[lane + 32] = ALPHA_C * bias[(size_t)b * NV + lane + 32];
  xv[lane]      = x_init[(size_t)b * NV + lane];
  xv[lane + 32] = x_init[(size_t)b * NV + lane + 32];
  __syncthreads();

  // ---- Newton-Schulz inversion on the matrix pipe: X <- X (2I - M X) ----
  for (int it = 0; it < NEWTON_IT; ++it) {
    // T = M * X
    for (int m = 0; m < 4; ++m) {
      v2f amk[16];
#pragma unroll
      for (int k = 0; k < 16; ++k) amk[k] = lds_afrag(Wm, m * 16, k * 4, lane);
#pragma unroll
      for (int n = 0; n < 4; ++n) {
        v8f acc = {};
#pragma unroll
        for (int k = 0; k < 16; ++k)
          acc = wmma_f32_16x16x4(amk[k], lds_bfrag(Wx, k * 4, n * 16, lane), acc);
        lds_store_ctile(Wt, m * 16, n * 16, lane, acc);
      }
    }
    __syncthreads();
    // X = 2X - X*T, in place per row-block (row m fully consumed before overwrite)
    for (int m = 0; m < 4; ++m) {
      v2f amk[16];
#pragma unroll
      for (int k = 0; k < 16; ++k) amk[k] = lds_afrag(Wx, m * 16, k * 4, lane);
#pragma unroll
      for (int n = 0; n < 4; ++n) {
        v8f acc = {};
#pragma unroll
        for (int k = 0; k < 16; ++k)
          acc = wmma_f32_16x16x4(amk[k], lds_bfrag(Wt, k * 4, n * 16, lane), acc);
        const int nc = n * 16 + (lane & 15);
        const int r0 = m * 16 + ((lane >> 4) << 3);
        v8f res;
#pragma unroll
        for (int v = 0; v < 8; ++v) res[v] = 2.0f * Wx[(r0 + v) * STR + nc] - acc[v];
        lds_store_ctile(Wx, m * 16, n * 16, lane, res);
      }
    }
    __syncthreads();
  }

  // ---- keep M^-1 resident as WMMA A-fragments (64 x v2f = 128 VGPRs/lane) ----
  v2f af[4][16];
#pragma unroll
  for (int m = 0; m < 4; ++m)
#pragma unroll
    for (int k = 0; k < 16; ++k)
      af[m][k] = lds_afrag(Wx, m * 16, k * 4, lane);

  // ---- Peaceman-Rachford fixed point ----
  const bool col0 = (lane & 15) == 0;   // only column 0 of each WMMA D-tile is used
  const int  hi16 = lane >> 4;          // 0: lanes 0-15, 1: lanes 16-31
  for (int iter = 0; iter < MAX_IT; ++iter) {
#pragma unroll
    for (int h = 0; h < 2; ++h) {
      const int i  = lane + 32 * h;
      const float xi = xv[i];
      const float z  = tanhf(xi);
      const float u  = 2.0f * z - xi;
      uh[i] = u;
      vv[i] = u + ab[i];
    }
    __syncthreads();
    // y = M^-1 * v : vector embedded in column 0 of the 4x16 B operand
#pragma unroll
    for (int m = 0; m < 4; ++m) {
      v8f acc = {};
#pragma unroll
      for (int k = 0; k < 16; ++k) {
        const float e0 = vv[4 * k + 2 * hi16];
        const float e1 = vv[4 * k + 2 * hi16 + 1];
        v2f bf;
        bf.x = col0 ? e0 : 0.0f;
        bf.y = col0 ? e1 : 0.0f;
        acc = wmma_f32_16x16x4(af[m][k], bf, acc);
      }
      if (col0) {
        const int base = m * 16 + hi16 * 8;
#pragma unroll
        for (int v = 0; v < 8; ++v) yn[base + v] = acc[v];
      }
    }
    __syncthreads();
    float d2 = 0.0f;
#pragma unroll
    for (int h = 0; h < 2; ++h) {
      const int i  = lane + 32 * h;
      const float xn = 2.0f * yn[i] - uh[i];
      const float dd = xn - xv[i];
      d2 += dd * dd;
      xv[i] = xn;
    }
    __syncthreads();
#pragma unroll
    for (int off = 16; off > 0; off >>= 1) d2 += __shfl_xor(d2, off, 32);
    if (d2 < 1.0e-12f) break;   // per-batch tol ~ TOL/sqrt(B), uniform branch
  }

  out[(size_t)b * NV + lane]      = xv[lane];
  out[(size_t)b * NV + lane + 32] = xv[lane + 32];
}

extern "C" void kernel_launch(void* const* d_in, const int* in_sizes, int n_in,
                              void* d_out, int out_size, void* d_ws, size_t ws_size,
                              hipStream_t stream) {
  const float* D11    = (const float*)d_in[0];
  const float* bias   = (const float*)d_in[1];
  const float* x_init = (const float*)d_in[2];
  float* out = (float*)d_out;
  const int B = in_sizes[0] / (NV * NV);   // 16384
  implicit_cell_pr_kernel<<<B, 32, 0, stream>>>(D11, bias, x_init, out);
}